// Loss_17695265260051
// MI455X (gfx1250) — compile-verified
//
#include <hip/hip_runtime.h>
#include <math.h>

typedef __attribute__((ext_vector_type(2))) float v2f;
typedef __attribute__((ext_vector_type(8))) float v8f;

#define BB 64
#define NN 1024
#define MM 1024
#define TILES (NN / 16) // 64 column tiles / row tiles

// workspace layout in floats
#define WS_COLMIN 0
#define WS_ROWSUM (2 * BB * MM)
#define WS_ASYM (WS_ROWSUM + 2 * BB)
#define WS_COLSUM (WS_ASYM + 2 * BB)
#define WS_TOTAL (WS_COLSUM + 2 * BB)

__device__ __forceinline__ float fast_sqrtf(float x) {
#if defined(__gfx1250__)
    return __builtin_amdgcn_sqrtf(x); // single v_sqrt_f32, ~1 ulp: fine for a loss
#else
    return sqrtf(x);
#endif
}

__global__ void init_ws_kernel(float* ws) {
    int i = blockIdx.x * blockDim.x + threadIdx.x;
    if (i < 2 * BB * MM) {
        ((unsigned int*)ws)[i] = 0x7F800000u; // +inf (colmin)
    } else if (i < WS_TOTAL) {
        ws[i] = 0.0f; // rowsum / asym / colsum accumulators
    }
}

// One block = one (pred, batch, 8 consecutive row tiles); one wave per row tile.
// Each wave runs 64 f32 WMMAs over the column tiles, keeping running min of d^2
// per row in registers and per column in a block-shared LDS table (ds_min_u32),
// flushed once per block to the L2-resident workspace. Target-tile loads are
// software-pipelined one tile ahead of the WMMA.
__global__ __launch_bounds__(256) void chamfer_wmma_kernel(
    const float* __restrict__ pred1, const float* __restrict__ pred2,
    const float* __restrict__ targ, float* __restrict__ ws) {
    __shared__ unsigned int colmin_s[MM]; // 4 KB: block-local column minima (d^2 as uint)

    const int lane = threadIdx.x & 31;
    const int waveInBlk = threadIdx.x >> 5;
    // blockIdx: 0..1023 -> pred (1b), batch (6b), row-tile-group (3b)
    const int pred = blockIdx.x >> 9;
    const int b = (blockIdx.x >> 3) & 63;
    const int rt = ((blockIdx.x & 7) << 3) + waveInBlk; // 0..63

    // init block-local colmin table
    for (int i = threadIdx.x; i < MM; i += 256) colmin_s[i] = 0x7F800000u; // +inf
    __syncthreads();

    const float* __restrict__ P = pred ? pred2 : pred1;
    const int li = lane & 15;
    const int hi = lane >> 4;

    // ---- Build A tile: 16 pred points, K=4 slot carries |a|^2 ----
    // f32 16x4 A layout: lanes 0-15 -> v0=K0(ax), v1=K1(ay);
    //                    lanes16-31 -> v0=K2(az), v1=K3(|a|^2)
    const float* ap = P + ((size_t)b * NN + (size_t)rt * 16 + li) * 3;
    const float px = ap[0], py = ap[1], pz = ap[2];
    const float n2 = px * px + py * py + pz * pz;
    v2f amat;
    amat.x = hi ? pz : px;
    amat.y = hi ? n2 : py;

    // ---- asym L2 term, hoisted out of the hot loop (diagonal pairing) ----
    const float* dp = targ + ((size_t)b * MM + (size_t)rt * 16 + li) * 3;
    float asym_part;
    {
        const float dx = px - dp[0], dy = py - dp[1], dz = pz - dp[2];
        asym_part = fast_sqrtf(dx * dx + dy * dy + dz * dz);
    }

    v8f rowmin;
#pragma unroll
    for (int v = 0; v < 8; ++v) rowmin[v] = 3.0e38f;

    // ---- software-pipelined column-tile loop ----
    const float* tp = targ + ((size_t)b * MM + li) * 3; // tile 0 for this lane
    float tx = tp[0], ty = tp[1], tz = tp[2];
    int sidx = li; // LDS colmin index, +16 per tile

    for (int ct = 0; ct < TILES; ++ct) {
        // issue next tile's load now (clamped pointer: no branch, no OOB)
        const float* tn = (ct < TILES - 1) ? (tp + 48) : tp;
        const float nx = tn[0], ny = tn[1], nz = tn[2];

        // ---- Build B tile: 16 target points, scaled by -2, K=3 row of 1s ----
        // B 4x16 layout: v0={K0 lanes0-15, K1 lanes16-31}, v1={K2, K3}
        const float b2 = tx * tx + ty * ty + tz * tz;
        v2f bmat;
        bmat.x = hi ? (-2.0f * ty) : (-2.0f * tx);
        bmat.y = hi ? 1.0f : (-2.0f * tz);

        v8f c = {};
#if defined(__gfx1250__)
        // D[n,m] = |a_n|^2 - 2 a_n . b_m   (add |b_m|^2 per lane below)
        c = __builtin_amdgcn_wmma_f32_16x16x4_f32(
            /*neg_a=*/false, amat, /*neg_b=*/false, bmat,
            /*c_mod=*/(short)0, c, /*reuse_a=*/false, /*reuse_b=*/false);
#endif
        float cmin = 3.0e38f;
#pragma unroll
        for (int v = 0; v < 8; ++v) {
            float d2 = c[v] + b2;       // column m = lane%16: this lane's |b|^2
            d2 = d2 > 0.0f ? d2 : 0.0f; // clamp fp round-off (also keeps uint order valid)
            rowmin[v] = fminf(rowmin[v], d2);
            cmin = fminf(cmin, d2);
        }
        // both half-waves (rows 0-7 and 8-15) combine into the block table:
        // non-negative floats order like uints -> ds_min_u32 is exact
        atomicMin(&colmin_s[sidx], __float_as_uint(cmin));
        sidx += 16;

        // rotate pipeline registers
        tx = nx; ty = ny; tz = nz;
        tp = tn;
    }

    // ---- Row minima: 16-lane min-reduce per VGPR, then sqrt and sum ----
    float rsum = 0.0f;
#pragma unroll
    for (int v = 0; v < 8; ++v) {
        float m = rowmin[v];
        m = fminf(m, __shfl_xor(m, 1, 32));
        m = fminf(m, __shfl_xor(m, 2, 32));
        m = fminf(m, __shfl_xor(m, 4, 32));
        m = fminf(m, __shfl_xor(m, 8, 32));
        rsum += fast_sqrtf(m);
    }
    rsum += __shfl_xor(rsum, 16, 32); // rows 0-7 + rows 8-15
    if (lane == 0) atomicAdd(&ws[WS_ROWSUM + pred * BB + b], rsum);

    // ---- asym: sum the 16 diagonal distances of this tile ----
    asym_part += __shfl_xor(asym_part, 1, 32);
    asym_part += __shfl_xor(asym_part, 2, 32);
    asym_part += __shfl_xor(asym_part, 4, 32);
    asym_part += __shfl_xor(asym_part, 8, 32);
    if (lane == 0) atomicAdd(&ws[WS_ASYM + pred * BB + b], asym_part);

    // ---- flush block-local column minima to global (8x fewer atomics) ----
    __syncthreads();
    unsigned int* __restrict__ colmin_g =
        (unsigned int*)(ws + WS_COLMIN) + (size_t)(pred * BB + b) * MM;
    for (int i = threadIdx.x; i < MM; i += 256) {
        atomicMin(&colmin_g[i], colmin_s[i]);
    }
}

// Reduce col minima: colsum[blk] = sum_m sqrt(colmin[blk][m]); blk = pred*B+b
__global__ __launch_bounds__(256) void colred_kernel(const float* __restrict__ colmin_ws,
                                                     float* __restrict__ colsum) {
    __shared__ float s[256];
    const int blk = blockIdx.x;
    const float* cm = colmin_ws + (size_t)blk * MM;
    float sum = 0.0f;
    for (int j = threadIdx.x; j < MM; j += 256) {
        sum += fast_sqrtf(fmaxf(cm[j], 0.0f));
    }
    s[threadIdx.x] = sum;
    __syncthreads();
    for (int off = 128; off > 0; off >>= 1) {
        if (threadIdx.x < off) s[threadIdx.x] += s[threadIdx.x + off];
        __syncthreads();
    }
    if (threadIdx.x == 0) colsum[blk] = s[0];
}

// Per-batch loss blend + mean over batches -> 3 scalars
__global__ __launch_bounds__(64) void final_kernel(const float* __restrict__ ws,
                                                   const float* __restrict__ flag1,
                                                   const float* __restrict__ flag2,
                                                   const float* __restrict__ symf,
                                                   float* __restrict__ out) {
    __shared__ float sAll[64], sL1[64], sL2[64];
    const int b = threadIdx.x; // 64 threads, one per batch
    const float* rowsum = ws + WS_ROWSUM;
    const float* asym = ws + WS_ASYM;
    const float* colsum = ws + WS_COLSUM;
    const float inv = 1.0f / (float)NN;

    const float sym1 = 0.5f * (rowsum[b] + colsum[b]) * inv;
    const float as1 = asym[b] * inv;
    const float sym2 = 0.5f * (rowsum[BB + b] + colsum[BB + b]) * inv;
    const float as2 = asym[BB + b] * inv;
    const float sf = symf[b];
    const float l1 = flag1[b] * (sf * sym1 + (1.0f - sf) * as1);
    const float l2 = flag2[b] * (sf * sym2 + (1.0f - sf) * as2);

    sAll[b] = l1 + l2;
    sL1[b] = l1;
    sL2[b] = l2;
    __syncthreads();
    for (int off = 32; off > 0; off >>= 1) {
        if (b < off) {
            sAll[b] += sAll[b + off];
            sL1[b] += sL1[b + off];
            sL2[b] += sL2[b + off];
        }
        __syncthreads();
    }
    if (b == 0) {
        out[0] = sAll[0] / (float)BB;
        out[1] = sL1[0] / (float)BB;
        out[2] = sL2[0] / (float)BB;
    }
}

extern "C" void kernel_launch(void* const* d_in, const int* in_sizes, int n_in,
                              void* d_out, int out_size, void* d_ws, size_t ws_size,
                              hipStream_t stream) {
    const float* p1 = (const float*)d_in[0]; // pred_points1 [B,N,3]
    const float* p2 = (const float*)d_in[1]; // pred_points2 [B,N,3]
    const float* tg = (const float*)d_in[2]; // target_points [B,M,3]
    const float* f1 = (const float*)d_in[3]; // flag1 [B]
    const float* f2 = (const float*)d_in[4]; // flag2 [B]
    const float* sf = (const float*)d_in[5]; // sym_flag [B]
    float* ws = (float*)d_ws;
    float* out = (float*)d_out;

    init_ws_kernel<<<(WS_TOTAL + 255) / 256, 256, 0, stream>>>(ws);
    // 2 preds * 64 batches * 8 row-tile groups = 1024 blocks of 8 waves
    chamfer_wmma_kernel<<<1024, 256, 0, stream>>>(p1, p2, tg, ws);
    colred_kernel<<<2 * BB, 256, 0, stream>>>(ws + WS_COLMIN, ws + WS_COLSUM);
    final_kernel<<<1, 64, 0, stream>>>(ws, f1, f2, sf, out);
}